// KLLoss_1967095021605
// MI455X (gfx1250) — compile-verified
//
#include <hip/hip_runtime.h>
#include <math.h>

typedef float v2f __attribute__((ext_vector_type(2)));
typedef float v8f __attribute__((ext_vector_type(8)));

#define CCLS 10

static __device__ __forceinline__ v2f ntload2(const float* p) {
  return __builtin_nontemporal_load((const v2f*)p);
}

// Each wave processes 16-row groups. For group g:
//   - WMMA f32 16x16x4 (x3, K=0..3 / 4..7 / 8..9+zero-pad) accumulates
//     acc += P16 * T16^T across ALL groups; diag(acc) = sum_g dot(t_i, p_i),
//     extracted once after the loop.
//   - Lanes L and L^16 jointly hold row (L&15): cols {0,1,4,5,8,9} in lanes
//     0-15, cols {2,3,6,7} in lanes 16-31. shfl_xor(16) completes the
//     row max / sum-exp / sum-t. Hi lanes use a -inf sentinel so the
//     softmax path is branchless (exp(-inf - m) == 0).
__global__ __launch_bounds__(256) void klloss_main_kernel(
    const float* __restrict__ P, const float* __restrict__ T,
    float* __restrict__ blockPartials, int nRows) {
  const int lane = threadIdx.x & 31;
  const int warp = threadIdx.x >> 5;
  const int wavesPerBlock = blockDim.x >> 5;
  const int gwave = blockIdx.x * wavesPerBlock + warp;
  const int nwaves = gridDim.x * wavesPerBlock;
  const int groups = nRows >> 4;            // 16 rows per group

  const int row16 = lane & 15;              // row within the 16-row tile
  const int hi    = lane >> 4;              // 0: K-low half, 1: K-high half
  const int colbase = 2 * hi;               // + 4*c per K-chunk
  const float loflag = (hi == 0) ? 1.0f : 0.0f;
  const float NEGINF = -__builtin_inff();

  float partial = 0.0f;
  v8f acc = {};                             // chained WMMA accumulator

  unsigned off = (unsigned)(gwave * 16 + row16) * CCLS + (unsigned)colbase;
  const unsigned stride = (unsigned)nwaves * 16u * CCLS;

  for (int g = gwave; g < groups; g += nwaves, off += stride) {
    // Exactly-once loads: 6 float2 for lanes 0-15, 4 float2 for lanes 16-31.
    v2f a0 = ntload2(P + off);
    v2f a1 = ntload2(P + off + 4);
    v2f b0 = ntload2(T + off);
    v2f b1 = ntload2(T + off + 4);
    v2f a2, b2, e2;
    if (hi == 0) {                          // cols 8,9 valid
      a2 = ntload2(P + off + 8);
      b2 = ntload2(T + off + 8);
      e2 = a2;
    } else {                                // cols 10,11 are K-padding
      a2 = (v2f){0.0f, 0.0f};               // zeros for WMMA (avoid -inf*0)
      b2 = (v2f){0.0f, 0.0f};
      e2 = (v2f){NEGINF, NEGINF};           // -inf sentinel for softmax
    }

    // acc += P * T^T over K=0..11 (last two K zero-padded).
    acc = __builtin_amdgcn_wmma_f32_16x16x4_f32(false, a0, false, b0,
                                                (short)0, acc, false, false);
    acc = __builtin_amdgcn_wmma_f32_16x16x4_f32(false, a1, false, b1,
                                                (short)0, acc, false, false);
    acc = __builtin_amdgcn_wmma_f32_16x16x4_f32(false, a2, false, b2,
                                                (short)0, acc, false, false);

    // Branchless log-sum-exp across the lane pair.
    float m = fmaxf(fmaxf(a0.x, a0.y), fmaxf(a1.x, a1.y));
    m = fmaxf(m, fmaxf(e2.x, e2.y));
    m = fmaxf(m, __shfl_xor(m, 16));

    float s = __expf(a0.x - m) + __expf(a0.y - m) +
              __expf(a1.x - m) + __expf(a1.y - m) +
              __expf(e2.x - m) + __expf(e2.y - m);
    s += __shfl_xor(s, 16);
    float logZ = m + __logf(s);

    float st = (b0.x + b0.y) + (b1.x + b1.y) + (b2.x + b2.y);
    st += __shfl_xor(st, 16);

    // Each row counted once (lanes 0-15); dot term handled via acc diag.
    partial = fmaf(st * logZ, loflag, partial);
  }

  // diag(acc): row m<=7 at lane m (VGPR m); row m>=8 at lane m+16 (VGPR m-8).
  {
    const int idx = lane & 7;
    float d = acc[0];
    d = (idx == 1) ? acc[1] : d;
    d = (idx == 2) ? acc[2] : d;
    d = (idx == 3) ? acc[3] : d;
    d = (idx == 4) ? acc[4] : d;
    d = (idx == 5) ? acc[5] : d;
    d = (idx == 6) ? acc[6] : d;
    d = (idx == 7) ? acc[7] : d;
    if ((lane < 8) || (lane >= 24)) partial -= d;
  }

  // Wave32 reduction, then 8-wave LDS reduction, one store per block.
  #pragma unroll
  for (int offr = 16; offr > 0; offr >>= 1) partial += __shfl_xor(partial, offr);

  __shared__ float wsum[8];
  if (lane == 0) wsum[warp] = partial;
  __syncthreads();
  if (threadIdx.x == 0) {
    float bsum = 0.0f;
    for (int i = 0; i < wavesPerBlock; ++i) bsum += wsum[i];
    blockPartials[blockIdx.x] = bsum;       // overwritten every call: deterministic
  }
}

__global__ __launch_bounds__(256) void klloss_finalize_kernel(
    const float* __restrict__ partials, int n, float* __restrict__ out,
    float invN) {
  float s = 0.0f;
  for (int i = threadIdx.x; i < n; i += blockDim.x) s += partials[i];
  #pragma unroll
  for (int off = 16; off > 0; off >>= 1) s += __shfl_xor(s, off);

  __shared__ float wsum[8];
  const int lane = threadIdx.x & 31;
  const int warp = threadIdx.x >> 5;
  if (lane == 0) wsum[warp] = s;
  __syncthreads();
  if (threadIdx.x == 0) {
    float t = 0.0f;
    const int nw = blockDim.x >> 5;
    for (int i = 0; i < nw; ++i) t += wsum[i];
    out[0] = t * invN;
  }
}

extern "C" void kernel_launch(void* const* d_in, const int* in_sizes, int n_in,
                              void* d_out, int out_size, void* d_ws, size_t ws_size,
                              hipStream_t stream) {
  const float* P = (const float*)d_in[0];   // predicted [N, 10] f32
  const float* T = (const float*)d_in[1];   // target    [N, 10] f32
  float* out = (float*)d_out;               // scalar f32
  float* partials = (float*)d_ws;

  const int nRows = in_sizes[0] / CCLS;     // N = 4,194,304

  int blocks = 2048;                        // 16384 waves; 16 groups each
  const int maxBlocks = (int)(ws_size / sizeof(float));
  if (blocks > maxBlocks) blocks = maxBlocks > 0 ? maxBlocks : 1;

  klloss_main_kernel<<<blocks, 256, 0, stream>>>(P, T, partials, nRows);
  klloss_finalize_kernel<<<1, 256, 0, stream>>>(partials, blocks, out,
                                                1.0f / (float)nRows);
}